// CompGCNConv_18734647345720
// MI455X (gfx1250) — compile-verified
//
#include <hip/hip_runtime.h>

typedef __attribute__((ext_vector_type(2))) float v2f;
typedef __attribute__((ext_vector_type(8))) float v8f;

#define DVEC   128     // D_IN == D_OUT
#define KSTEPS 32      // 128 / 4 (K per WMMA)
#define NTILES 8       // 128 / 16 (N tiles)

// LDS layout: bw4[s][n] = { W[n][4s], W[n][4s+1], W[n][4s+2], W[n][4s+3] }
// B-fragment for k-step s, lane (h = lane>>4, m = lane&15), tile j:
//   float2 at &bw4[s][j*16+m] + h*2 floats -> conflict-free ds_load_b64
//   (lanes 0-15 hit even bank pairs, lanes 16-31 odd pairs).

__device__ __forceinline__ void load_w_to_lds(float4* bw4, const float* __restrict__ W) {
    const float4* W4 = (const float4*)W;           // W is (128,128) row-major
    for (int idx = threadIdx.x; idx < KSTEPS * DVEC; idx += blockDim.x) {
        int n = idx >> 5;        // output channel
        int s = idx & 31;        // k-step
        bw4[s * DVEC + n] = W4[idx];               // coalesced global read
    }
}

// Out[row] = X[row] @ W^T (+ bias), full overwrite. Used for self-transform and rel update.
__global__ __launch_bounds__(256) void compgcn_dense(const float* __restrict__ X,
                                                     const float* __restrict__ W,
                                                     const float* __restrict__ bias,
                                                     float* __restrict__ Out,
                                                     int nrows) {
    __shared__ float4 bw4[KSTEPS * DVEC];          // 64 KB
    load_w_to_lds(bw4, W);
    __syncthreads();

    const int lane = threadIdx.x & 31;
    const int m    = lane & 15;                    // row within tile
    const int h    = lane >> 4;                    // K half-pair selector
    const int wave = threadIdx.x >> 5;
    const int wavesPerBlock = blockDim.x >> 5;
    const int gwave = blockIdx.x * wavesPerBlock + wave;
    const int nwaves = gridDim.x * wavesPerBlock;
    const int ntileRows = (nrows + 15) >> 4;
    const float* bwl = (const float*)bw4 + m * 4 + 2 * h;   // per-lane LDS base

    // Tile-invariant bias fragment: one value per n-tile for this lane's column.
    float bv[NTILES];
    #pragma unroll
    for (int j = 0; j < NTILES; ++j) bv[j] = bias ? bias[j * 16 + m] : 0.f;

    for (int t = gwave; t < ntileRows; t += nwaves) {
        int row = t * 16 + m;
        int rc  = row < nrows ? row : (nrows - 1);
        const float* xrow = X + (size_t)rc * DVEC;

        v8f acc[NTILES];
        #pragma unroll
        for (int j = 0; j < NTILES; ++j) acc[j] = (v8f){0.f,0.f,0.f,0.f,0.f,0.f,0.f,0.f};

        #pragma unroll 2
        for (int s = 0; s < KSTEPS; ++s) {
            // Issue all fetches first so DS/LOAD counters drain while WMMAs execute.
            v2f a = *(const v2f*)(xrow + 4 * s + 2 * h);
            v2f b[NTILES];
            #pragma unroll
            for (int j = 0; j < NTILES; ++j)
                b[j] = *(const v2f*)(bwl + s * DVEC * 4 + j * 64);
            #pragma unroll
            for (int j = 0; j < NTILES; ++j)
                acc[j] = __builtin_amdgcn_wmma_f32_16x16x4_f32(
                    false, a, false, b[j], (short)0, acc[j], false, false);
        }

        // C layout: vgpr i of tile j holds (row = t*16 + i + 8h, col = j*16 + m).
        if (t * 16 + 16 <= nrows) {
            // Full tile: unconditional coalesced stores (no exec juggling).
            #pragma unroll
            for (int j = 0; j < NTILES; ++j) {
                float* orow = Out + (size_t)(t * 16 + 8 * h) * DVEC + j * 16 + m;
                #pragma unroll
                for (int i = 0; i < 8; ++i)
                    orow[(size_t)i * DVEC] = acc[j][i] + bv[j];
            }
        } else {
            // Tail tile only.
            #pragma unroll
            for (int j = 0; j < NTILES; ++j) {
                #pragma unroll
                for (int i = 0; i < 8; ++i) {
                    int grow = t * 16 + i + 8 * h;
                    if (grow < nrows)
                        Out[(size_t)grow * DVEC + j * 16 + m] = acc[j][i] + bv[j];
                }
            }
        }
    }
}

// Per 16-edge tile: acc = ((x[src] - rel[etype]) * w) @ W_fwd^T, scatter-atomic into Out[dst].
__global__ __launch_bounds__(256) void compgcn_edges(const float* __restrict__ X,
                                                     const float* __restrict__ Rel,
                                                     const int* __restrict__ src,
                                                     const int* __restrict__ dst,
                                                     const int* __restrict__ etype,
                                                     const float* __restrict__ ew,
                                                     const float* __restrict__ W,
                                                     float* __restrict__ Out,
                                                     int nedges) {
    __shared__ float4 bw4[KSTEPS * DVEC];          // 64 KB
    load_w_to_lds(bw4, W);
    __syncthreads();

    const int lane = threadIdx.x & 31;
    const int m    = lane & 15;
    const int h    = lane >> 4;
    const int wave = threadIdx.x >> 5;
    const int wavesPerBlock = blockDim.x >> 5;
    const int gwave = blockIdx.x * wavesPerBlock + wave;
    const int nwaves = gridDim.x * wavesPerBlock;
    const int ntiles = (nedges + 15) >> 4;
    const float* bwl = (const float*)bw4 + m * 4 + 2 * h;

    for (int t = gwave; t < ntiles; t += nwaves) {
        int e  = t * 16 + m;
        bool valid = e < nedges;
        int ec = valid ? e : 0;
        float w  = valid ? ew[ec] : 0.f;           // w=0 -> exact-zero contribution for tail rows
        int   sN = src[ec];
        int   ty = etype[ec];
        int   dN = dst[ec];
        const float* xrow = X   + (size_t)sN * DVEC;
        const float* rrow = Rel + (size_t)ty * DVEC;

        v8f acc[NTILES];
        #pragma unroll
        for (int j = 0; j < NTILES; ++j) acc[j] = (v8f){0.f,0.f,0.f,0.f,0.f,0.f,0.f,0.f};

        #pragma unroll 2
        for (int s = 0; s < KSTEPS; ++s) {
            // Issue the two gathers and all 8 LDS B-fetches before any consumption.
            v2f xv = *(const v2f*)(xrow + 4 * s + 2 * h);    // L2-resident gather
            v2f rv = *(const v2f*)(rrow + 4 * s + 2 * h);
            v2f b[NTILES];
            #pragma unroll
            for (int j = 0; j < NTILES; ++j)
                b[j] = *(const v2f*)(bwl + s * DVEC * 4 + j * 64);
            v2f a;
            a.x = (xv.x - rv.x) * w;                         // fold edge weight into A (linear)
            a.y = (xv.y - rv.y) * w;
            #pragma unroll
            for (int j = 0; j < NTILES; ++j)
                acc[j] = __builtin_amdgcn_wmma_f32_16x16x4_f32(
                    false, a, false, b[j], (short)0, acc[j], false, false);
        }

        // Scatter-add: vgpr i of the C tile holds row (i + 8h); fetch that edge's dst via shuffle.
        #pragma unroll
        for (int i = 0; i < 8; ++i) {
            int r  = i + 8 * h;
            int dd = __shfl(dN, r, 32);                      // lane r (0-15) holds row r's dst
            float* orow = Out + (size_t)dd * DVEC;
            #pragma unroll
            for (int j = 0; j < NTILES; ++j)
                atomicAdd(orow + j * 16 + m, acc[j][i]);     // global_atomic_add_f32
        }
    }
}

extern "C" void kernel_launch(void* const* d_in, const int* in_sizes, int n_in,
                              void* d_out, int out_size, void* d_ws, size_t ws_size,
                              hipStream_t stream) {
    const float* x         = (const float*)d_in[0];
    const int*   edge_idx  = (const int*)d_in[1];   // (2, E)
    const int*   edge_type = (const int*)d_in[2];   // (E,)
    const float* rel_emb   = (const float*)d_in[3]; // (R, 128)
    const float* edge_w    = (const float*)d_in[4]; // (E,)
    const float* W_self    = (const float*)d_in[5];
    const float* W_forward = (const float*)d_in[6];
    const float* W_rel     = (const float*)d_in[7];
    const float* bias      = (const float*)d_in[8];

    const int N = in_sizes[0] / DVEC;   // 100000 nodes
    const int E = in_sizes[2];          // 625000 edges
    const int R = in_sizes[3] / DVEC;   // 200 relations

    const int* src = edge_idx;
    const int* dst = edge_idx + E;

    float* out_nodes = (float*)d_out;
    float* out_rel   = out_nodes + (size_t)N * DVEC;

    // 1) Self transform: fully overwrites node output region (clean base for atomics every call).
    {
        int tiles  = (N + 15) >> 4;
        int blocks = (tiles + 7) >> 3;                      // 8 waves per 256-thread block
        compgcn_dense<<<blocks, 256, 0, stream>>>(x, W_self, bias, out_nodes, N);
    }
    // 2) Relation embedding update (tiny).
    {
        int tiles  = (R + 15) >> 4;
        int blocks = (tiles + 7) >> 3;
        compgcn_dense<<<blocks, 256, 0, stream>>>(rel_emb, W_rel, nullptr, out_rel, R);
    }
    // 3) Edge message GEMM + scatter-add (dominant cost).
    {
        int tiles  = (E + 15) >> 4;
        int blocks = (tiles + 7) >> 3;
        if (blocks > 2048) blocks = 2048;                   // grid-stride; amortize LDS W staging
        compgcn_edges<<<blocks, 256, 0, stream>>>(x, rel_emb, src, dst, edge_type, edge_w,
                                                  W_forward, out_nodes, E);
    }
}